// GAT_420906795031
// MI455X (gfx1250) — compile-verified
//
#include <hip/hip_runtime.h>
#include <hip/hip_bf16.h>

typedef float v2f __attribute__((ext_vector_type(2)));
typedef float v8f __attribute__((ext_vector_type(8)));

#define NEG_SLOPE 0.2f

// ---------------------------------------------------------------- utilities
__global__ void fill_k(float* __restrict__ p, long long n, float v) {
  long long i = (long long)blockIdx.x * blockDim.x + threadIdx.x;
  if (i < n) p[i] = v;
}

__device__ __forceinline__ void atomicMaxF(float* addr, float v) {
  // Sign-aware ordered-int trick: exact float max with hardware int atomics.
  if (v >= 0.0f) atomicMax((int*)addr, __float_as_int(v));
  else           atomicMin((unsigned int*)addr, __float_as_uint(v));
}

__device__ __forceinline__ void edge_sd(const long long* __restrict__ ei,
                                        int E, int e, int* s, int* d) {
  if (e < E) { *s = (int)ei[e]; *d = (int)ei[(long long)E + e]; }
  else       { int n = e - E; *s = n; *d = n; }   // appended self-loop
}

// ---------------------------------------------------------------- GEMM (WMMA + LDS)
// C[M,N] = A[M,K] @ B[K,N], row-major. Workgroup = 8 waves covering 128 rows x
// one 64-col block; K is processed in 128-deep blocks staged in LDS.
//
// LDS layout (pair-interleaved + padded):
//   pair-row p (= K rows 2p, 2p+1 of the block), stride 160 dwords:
//     Bs[p*160 + 2*c + 0] = B[2p  ][c],  Bs[p*160 + 2*c + 1] = B[2p+1][c]
//   -> a B fragment component pair is ONE ds_load_b64 straight into an even
//      VGPR pair (no shuffle moves), and the 160-dword stride puts the
//      half=1 lanes on banks 32..63 (half=0 on 0..31): conflict-free.
// Requires M%16==0, N%64==0, K%128==0.
#define KB 128
#define PSTR 160            // dwords per pair-row: 128 data + 32 pad

__global__ __launch_bounds__(256) void gemm_lds_k(const float* __restrict__ A,
                                                  const float* __restrict__ B,
                                                  float* __restrict__ C,
                                                  int M, int K, int N) {
  __shared__ float Bs[(KB / 2) * PSTR];   // 40 KB
  int tilesN = N >> 6;
  int rg = blockIdx.x / tilesN;
  int n0 = (blockIdx.x - rg * tilesN) << 6;

  int wave = threadIdx.x >> 5;
  int lane = threadIdx.x & 31;
  int m0 = (rg * 8 + wave) << 4;
  bool active = (m0 < M);                 // wave-uniform
  int row  = lane & 15;
  int half = lane >> 4;                   // 0: lanes 0-15, 1: lanes 16-31

  const float* Arow = A + (long long)(m0 + row) * K + 2 * half;
  const float* Bl   = Bs + half * PSTR + row * 2;

  v8f c0 = {}, c1 = {}, c2 = {}, c3 = {};

  for (int kb = 0; kb < K; kb += KB) {
    if (kb) __syncthreads();              // all waves done reading previous slab
    // cooperative pair-interleaved fill of the KB x 64 slab at column n0
    for (int i = threadIdx.x; i < (KB / 2) * 16; i += 256) {
      int p = i >> 4, c4 = (i & 15) << 2;           // pair-row, col group of 4
      const float* Bq = B + (long long)(kb + 2 * p) * N + n0 + c4;
      float4 r0 = *(const float4*)Bq;
      float4 r1 = *(const float4*)(Bq + N);
      float* o = Bs + p * PSTR + c4 * 2;
      *(float4*)(o)     = make_float4(r0.x, r1.x, r0.y, r1.y);
      *(float4*)(o + 4) = make_float4(r0.z, r1.z, r0.w, r1.w);
    }
    __syncthreads();
    if (!active) continue;                // wave-uniform skip; EXEC stays all-1s

    // A 16x4 frag: lanes 0-15 hold K=k,k+1; lanes 16-31 hold K=k+2,k+3 (b64)
    v2f a_cur; a_cur.x = Arow[kb]; a_cur.y = Arow[kb + 1];
#pragma unroll 4
    for (int t = 0; t < KB / 4; ++t) {
      v2f a_nxt = a_cur;
      if (t + 1 < KB / 4) { a_nxt.x = Arow[kb + 4 * t + 4]; a_nxt.y = Arow[kb + 4 * t + 5]; }
      const float* Bp = Bl + 2 * t * PSTR;
      // B 4x16 frag j: pair-row 2t+half, cols 16*j+row -> single b64 each
      v2f b0 = *(const v2f*)(Bp);
      v2f b1 = *(const v2f*)(Bp + 32);
      v2f b2 = *(const v2f*)(Bp + 64);
      v2f b3 = *(const v2f*)(Bp + 96);
      c0 = __builtin_amdgcn_wmma_f32_16x16x4_f32(false, a_cur, false, b0, (short)0, c0, false, false);
      c1 = __builtin_amdgcn_wmma_f32_16x16x4_f32(false, a_cur, false, b1, (short)0, c1, false, false);
      c2 = __builtin_amdgcn_wmma_f32_16x16x4_f32(false, a_cur, false, b2, (short)0, c2, false, false);
      c3 = __builtin_amdgcn_wmma_f32_16x16x4_f32(false, a_cur, false, b3, (short)0, c3, false, false);
      a_cur = a_nxt;
    }
  }
  if (!active) return;
  // C/D layout: VGPR v, lanes 0-15 -> M=m0+v, lanes 16-31 -> M=m0+v+8; N=16*j+row
  for (int v = 0; v < 8; ++v) {
    long long r = m0 + v + 8 * half;
    float* Cr = C + r * (long long)N + n0 + row;
    Cr[0] = c0[v]; Cr[16] = c1[v]; Cr[32] = c2[v]; Cr[48] = c3[v];
  }
}

// ------------------------------------------------------- per-node attention dots
__global__ void alpha_k(const float* __restrict__ Hm, const float* __restrict__ a_s,
                        const float* __restrict__ a_d, float* __restrict__ AS,
                        float* __restrict__ AD, int NH, int H, int C) {
  int i = blockIdx.x * blockDim.x + threadIdx.x;
  if (i >= NH) return;
  int n = i / H, h = i % H;
  const float* hp = Hm + (long long)n * H * C + h * C;
  const float* as = a_s + h * C;
  const float* ad = a_d + h * C;
  float s = 0.f, d = 0.f;
  for (int c = 0; c < C; ++c) { float v = hp[c]; s += v * as[c]; d += v * ad[c]; }
  AS[i] = s; AD[i] = d;
}

// ------------------------------------------------------- edge passes
__global__ void edge_max_k(const long long* __restrict__ ei, int E, int Etot,
                           const float* __restrict__ AS, const float* __restrict__ AD,
                           float* __restrict__ Mx, int H) {
  int i = blockIdx.x * blockDim.x + threadIdx.x;
  if (i >= Etot * H) return;
  int e = i / H, h = i - e * H;
  int s, d; edge_sd(ei, E, e, &s, &d);
  float el = AS[s * H + h] + AD[d * H + h];
  el = el > 0.f ? el : NEG_SLOPE * el;
  atomicMaxF(&Mx[d * H + h], el);
}

__global__ void edge_denom_k(const long long* __restrict__ ei, int E, int Etot,
                             const float* __restrict__ AS, const float* __restrict__ AD,
                             const float* __restrict__ Mx, float* __restrict__ Dn, int H) {
  int i = blockIdx.x * blockDim.x + threadIdx.x;
  if (i >= Etot * H) return;
  int e = i / H, h = i - e * H;
  int s, d; edge_sd(ei, E, e, &s, &d);
  float el = AS[s * H + h] + AD[d * H + h];
  el = el > 0.f ? el : NEG_SLOPE * el;
  atomicAdd(&Dn[d * H + h], expf(el - Mx[d * H + h]));
}

// Precompute per-(edge,head) softmax weight so the scatter pass is pure traffic.
__global__ void edge_alpha_k(const long long* __restrict__ ei, int E, int Etot,
                             const float* __restrict__ AS, const float* __restrict__ AD,
                             const float* __restrict__ Mx, const float* __restrict__ Dn,
                             float* __restrict__ ALPHA, int H) {
  int i = blockIdx.x * blockDim.x + threadIdx.x;
  if (i >= Etot * H) return;
  int e = i / H, h = i - e * H;
  int s, d; edge_sd(ei, E, e, &s, &d);
  float el = AS[s * H + h] + AD[d * H + h];
  el = el > 0.f ? el : NEG_SLOPE * el;
  ALPHA[i] = expf(el - Mx[d * H + h]) / (Dn[d * H + h] + 1e-16f);
}

// float4 gather of h[src], alpha-scale, 4x f32 atomic scatter-add into agg[dst].
// Entire working set (h, agg, edges, alpha) is L2-resident (~150MB < 192MB L2).
__global__ void edge_scatter_k(const long long* __restrict__ ei, int E,
                               const float* __restrict__ ALPHA,
                               const float* __restrict__ Hm, float* __restrict__ AGG,
                               int H, int C, int total) {
  int idx = blockIdx.x * blockDim.x + threadIdx.x;
  if (idx >= total) return;
  int q4 = (H * C) >> 2;                 // float4 slots per edge
  int e = idx / q4, q = idx - e * q4;
  int s, d; edge_sd(ei, E, e, &s, &d);
  float alpha = ALPHA[e * H + ((q << 2) / C)];
  float4 v = *((const float4*)(Hm + (long long)s * H * C) + q);
  float* ag = AGG + (long long)d * H * C + (q << 2);
  atomicAdd(ag + 0, v.x * alpha);
  atomicAdd(ag + 1, v.y * alpha);
  atomicAdd(ag + 2, v.z * alpha);
  atomicAdd(ag + 3, v.w * alpha);
}

// ------------------------------------------------------- epilogues
__global__ void bias_act_k(float* __restrict__ X, const float* __restrict__ b,
                           long long total, int F, int do_relu) {
  long long i = (long long)blockIdx.x * blockDim.x + threadIdx.x;
  if (i >= total) return;
  float v = X[i] + b[i % F];
  X[i] = do_relu ? (v > 0.f ? v : 0.f) : v;
}

// ---------------------------------------------------------------- launcher
extern "C" void kernel_launch(void* const* d_in, const int* in_sizes, int n_in,
                              void* d_out, int out_size, void* d_ws, size_t ws_size,
                              hipStream_t stream) {
  const float*     x      = (const float*)d_in[0];
  const long long* ei     = (const long long*)d_in[1];
  const float*     W1     = (const float*)d_in[2];
  const float*     a_src1 = (const float*)d_in[3];
  const float*     a_dst1 = (const float*)d_in[4];
  const float*     b1     = (const float*)d_in[5];
  const float*     W2     = (const float*)d_in[6];
  const float*     a_src2 = (const float*)d_in[7];
  const float*     a_dst2 = (const float*)d_in[8];
  const float*     b2     = (const float*)d_in[9];
  float* out = (float*)d_out;

  const int Fin = 128, H1 = 4, C = 64, F1 = H1 * C;   // 256
  const int N = in_sizes[0] / Fin;
  const int E = in_sizes[1] / 2;
  const int Etot = E + N;                              // with self-loops

  // workspace layout (floats)
  float* ws     = (float*)d_ws;
  float* h1     = ws;                        // N*256
  float* agg1   = h1   + (long long)N * F1;  // N*256 (becomes layer-1 output)
  float* h2     = agg1 + (long long)N * F1;  // N*64
  float* AS1    = h2   + (long long)N * C;   // N*4
  float* AD1    = AS1  + (long long)N * H1;
  float* M1     = AD1  + (long long)N * H1;
  float* D1     = M1   + (long long)N * H1;
  float* AS2    = D1   + (long long)N * H1;  // N
  float* AD2    = AS2  + N;
  float* M2     = AD2  + N;
  float* D2     = M2   + N;
  float* ALPHA1 = D2   + N;                  // Etot*4
  float* ALPHA2 = ALPHA1 + (long long)Etot * H1;  // Etot
  (void)ws_size; (void)n_in; (void)out_size;

  const float NEG_INF = -__builtin_huge_valf();
  auto nb = [](long long n, int t) { return (unsigned)((n + t - 1) / t); };

  // init accumulators
  fill_k<<<nb((long long)N * F1, 256), 256, 0, stream>>>(agg1, (long long)N * F1, 0.f);
  fill_k<<<nb((long long)N * H1, 256), 256, 0, stream>>>(M1, (long long)N * H1, NEG_INF);
  fill_k<<<nb((long long)N * H1, 256), 256, 0, stream>>>(D1, (long long)N * H1, 0.f);
  fill_k<<<nb(N, 256), 256, 0, stream>>>(M2, N, NEG_INF);
  fill_k<<<nb(N, 256), 256, 0, stream>>>(D2, N, 0.f);
  fill_k<<<nb((long long)N * C, 256), 256, 0, stream>>>(out, (long long)N * C, 0.f);

  const int rowGroups = (N / 16 + 7) / 8;   // 8 waves (128 rows) per workgroup

  // ---- layer 1 ----
  gemm_lds_k<<<(unsigned)(rowGroups * (F1 / 64)), 256, 0, stream>>>(x, W1, h1, N, Fin, F1);
  alpha_k<<<nb((long long)N * H1, 256), 256, 0, stream>>>(h1, a_src1, a_dst1, AS1, AD1,
                                                          N * H1, H1, C);
  edge_max_k<<<nb((long long)Etot * H1, 256), 256, 0, stream>>>(ei, E, Etot, AS1, AD1, M1, H1);
  edge_denom_k<<<nb((long long)Etot * H1, 256), 256, 0, stream>>>(ei, E, Etot, AS1, AD1, M1, D1, H1);
  edge_alpha_k<<<nb((long long)Etot * H1, 256), 256, 0, stream>>>(ei, E, Etot, AS1, AD1, M1, D1,
                                                                  ALPHA1, H1);
  {
    int total = Etot * (F1 / 4);
    edge_scatter_k<<<nb(total, 256), 256, 0, stream>>>(ei, E, ALPHA1, h1, agg1, H1, C, total);
  }
  bias_act_k<<<nb((long long)N * F1, 256), 256, 0, stream>>>(agg1, b1, (long long)N * F1, F1, 1);

  // ---- layer 2 ----
  gemm_lds_k<<<(unsigned)(rowGroups * (C / 64)), 256, 0, stream>>>(agg1, W2, h2, N, F1, C);
  alpha_k<<<nb(N, 256), 256, 0, stream>>>(h2, a_src2, a_dst2, AS2, AD2, N, 1, C);
  edge_max_k<<<nb(Etot, 256), 256, 0, stream>>>(ei, E, Etot, AS2, AD2, M2, 1);
  edge_denom_k<<<nb(Etot, 256), 256, 0, stream>>>(ei, E, Etot, AS2, AD2, M2, D2, 1);
  edge_alpha_k<<<nb(Etot, 256), 256, 0, stream>>>(ei, E, Etot, AS2, AD2, M2, D2, ALPHA2, 1);
  {
    int total = Etot * (C / 4);
    edge_scatter_k<<<nb(total, 256), 256, 0, stream>>>(ei, E, ALPHA2, h2, out, 1, C, total);
  }
  bias_act_k<<<nb((long long)N * C, 256), 256, 0, stream>>>(out, b2, (long long)N * C, C, 0);
}